// AbstractSpamClassifierWithTaggerIndexMapperAndDynamicGraph_73280732004547
// MI455X (gfx1250) — compile-verified
//
#include <hip/hip_runtime.h>

typedef __attribute__((ext_vector_type(2))) float v2f;
typedef __attribute__((ext_vector_type(8))) float v8f;

constexpr int Bn = 256;   // batch
constexpr int Sn = 512;   // sequence length
constexpr int En = 300;   // embedding dim
constexpr int Hn = 512;   // hidden dim
constexpr int H4 = 2048;  // 4*H (gates)

// f32 WMMA: D(16x16) = A(16x4) * B(4x16) + C, all f32 (matches reference numerics)
#define WMMA_F32(a, b, c) \
  __builtin_amdgcn_wmma_f32_16x16x4_f32(false, (a), false, (b), (short)0, (c), false, false)

__device__ __forceinline__ float sigmoidf(float v) {
  return 1.0f / (1.0f + __expf(-v));
}

// One recurrence step, fully fused:
//   gates = emb[x[:,s]] @ W_ih[tag]^T + h_in @ W_hh[tag]^T + b_ih[tag] + b_hh[tag]
//   c = sig(f)*c + sig(i)*tanh(g);  h_out = sig(o)*tanh(c)
// Grid: (Bn/64, Hn/64); block: 128 threads = 4 waves; each wave: 64M x 16N x 4 gates.
__global__ __launch_bounds__(128) void lstm_step_kernel(
    const int* __restrict__ x, const int* __restrict__ tags,
    const float* __restrict__ emb, const float* __restrict__ W_ih,
    const float* __restrict__ W_hh, const float* __restrict__ b_ih,
    const float* __restrict__ b_hh, const float* __restrict__ h_in,
    float* __restrict__ h_out, float* __restrict__ c, int s) {
  const int lane   = threadIdx.x & 31;
  const int wave   = threadIdx.x >> 5;
  const int laneHi = lane >> 4;          // 0: K{0,1} half, 1: K{2,3} half
  const int l15    = lane & 15;
  const int kofs   = laneHi * 2;

  const int m0 = blockIdx.x * 64;                 // batch tile base
  const int n0 = blockIdx.y * 64 + wave * 16;     // hidden-col tile base (per wave)

  const int tag = tags[s];
  const float* Wih_t = W_ih + (long)tag * H4 * En;
  const float* Whh_t = W_hh + (long)tag * H4 * Hn;

  // 4 M-subtiles x 4 gates of 16x16 f32 accumulators
  v8f acc[4][4];
#pragma unroll
  for (int mi = 0; mi < 4; ++mi)
#pragma unroll
    for (int g = 0; g < 4; ++g) acc[mi][g] = (v8f){};

  // A-matrix rows handled by this lane (M = m0 + mi*16 + (lane&15))
  int arow[4];
  const float* aemb[4];
  const float* ahid[4];
#pragma unroll
  for (int mi = 0; mi < 4; ++mi) {
    arow[mi] = m0 + mi * 16 + l15;
    const int eidx = x[arow[mi] * Sn + s];           // embedding gather index
    aemb[mi] = emb + (long)eidx * En + kofs;
    ahid[mi] = h_in + (long)arow[mi] * Hn + kofs;
  }

  // B-matrix: B[k][n] = W[gate*H + n][k] (row-major W -> contiguous float2 in k)
  const int bcol = n0 + l15;
  const float* bih_p[4];
  const float* bhh_p[4];
#pragma unroll
  for (int g = 0; g < 4; ++g) {
    bih_p[g] = Wih_t + (long)(g * Hn + bcol) * En + kofs;
    bhh_p[g] = Whh_t + (long)(g * Hn + bcol) * Hn + kofs;
  }

  // ---- Input projection: K = E = 300 (75 k-steps of 4) ----
#pragma unroll 2
  for (int k = 0; k < En; k += 4) {
    v2f a[4], bb[4];
#pragma unroll
    for (int mi = 0; mi < 4; ++mi) a[mi] = *(const v2f*)(aemb[mi] + k);
#pragma unroll
    for (int g = 0; g < 4; ++g) bb[g] = *(const v2f*)(bih_p[g] + k);
#pragma unroll
    for (int mi = 0; mi < 4; ++mi)
#pragma unroll
      for (int g = 0; g < 4; ++g) acc[mi][g] = WMMA_F32(a[mi], bb[g], acc[mi][g]);
  }

  // ---- Recurrent projection: K = H = 512 (128 k-steps of 4) ----
#pragma unroll 2
  for (int k = 0; k < Hn; k += 4) {
    v2f a[4], bb[4];
#pragma unroll
    for (int mi = 0; mi < 4; ++mi) a[mi] = *(const v2f*)(ahid[mi] + k);
#pragma unroll
    for (int g = 0; g < 4; ++g) bb[g] = *(const v2f*)(bhh_p[g] + k);
#pragma unroll
    for (int mi = 0; mi < 4; ++mi)
#pragma unroll
      for (int g = 0; g < 4; ++g) acc[mi][g] = WMMA_F32(a[mi], bb[g], acc[mi][g]);
  }

  // ---- Fused LSTM cell update (per-lane column bias is uniform over rows) ----
  float bsum[4];
#pragma unroll
  for (int g = 0; g < 4; ++g) {
    const int j = g * Hn + bcol;
    bsum[g] = b_ih[tag * H4 + j] + b_hh[tag * H4 + j];
  }

#pragma unroll
  for (int mi = 0; mi < 4; ++mi) {
#pragma unroll
    for (int r = 0; r < 8; ++r) {
      // C/D layout: VGPR r holds M = r (lanes 0-15) or M = r+8 (lanes 16-31)
      const int row = m0 + mi * 16 + r + laneHi * 8;
      const long idx = (long)row * Hn + bcol;
      const float gi = acc[mi][0][r] + bsum[0];
      const float gf = acc[mi][1][r] + bsum[1];
      const float gg = acc[mi][2][r] + bsum[2];
      const float go = acc[mi][3][r] + bsum[3];
      const float cn = sigmoidf(gf) * c[idx] + sigmoidf(gi) * tanhf(gg);
      c[idx] = cn;
      h_out[idx] = sigmoidf(go) * tanhf(cn);
    }
  }
}

// Final: out[b] = sigmoid(h[b,:] . fc_W + fc_b); copy h and c into d_out.
__global__ __launch_bounds__(256) void finalize_kernel(
    const float* __restrict__ h, const float* __restrict__ c,
    const float* __restrict__ fc_W, const float* __restrict__ fc_b,
    float* __restrict__ out) {
  const int i = blockIdx.x * 256 + threadIdx.x;
  if (i < Bn * Hn) {
    out[Bn + i] = h[i];
    out[Bn + Bn * Hn + i] = c[i];
  }
  if (i < Bn) {
    float acc = 0.0f;
    for (int k = 0; k < Hn; ++k) acc += h[i * Hn + k] * fc_W[k];
    out[i] = sigmoidf(acc + fc_b[0]);
  }
}

extern "C" void kernel_launch(void* const* d_in, const int* in_sizes, int n_in,
                              void* d_out, int out_size, void* d_ws, size_t ws_size,
                              hipStream_t stream) {
  const int*   x    = (const int*)d_in[0];
  const int*   tags = (const int*)d_in[1];
  const float* emb  = (const float*)d_in[2];
  const float* Wih  = (const float*)d_in[3];
  const float* Whh  = (const float*)d_in[4];
  const float* bih  = (const float*)d_in[5];
  const float* bhh  = (const float*)d_in[6];
  const float* fcW  = (const float*)d_in[7];
  const float* fcb  = (const float*)d_in[8];
  const float* h0   = (const float*)d_in[9];
  const float* c0   = (const float*)d_in[10];
  float* out = (float*)d_out;

  float* ws = (float*)d_ws;
  float* hb[2] = {ws, ws + (size_t)Bn * Hn};     // ping-pong hidden state
  float* cb = ws + 2 * (size_t)Bn * Hn;          // cell state (in place)

  hipMemcpyAsync(hb[0], h0, sizeof(float) * Bn * Hn, hipMemcpyDeviceToDevice, stream);
  hipMemcpyAsync(cb,    c0, sizeof(float) * Bn * Hn, hipMemcpyDeviceToDevice, stream);

  const dim3 grid(Bn / 64, Hn / 64);  // 4 x 8 blocks, 128 threads each
  for (int s = 0; s < Sn; ++s) {
    lstm_step_kernel<<<grid, 128, 0, stream>>>(x, tags, emb, Wih, Whh, bih, bhh,
                                               hb[s & 1], hb[(s + 1) & 1], cb, s);
  }
  // After 512 steps (even), final h lives in hb[0].
  finalize_kernel<<<(Bn * Hn) / 256, 256, 0, stream>>>(hb[0], cb, fcW, fcb, out);
}